// ParallelDeformableCrossAttention_68178310857199
// MI455X (gfx1250) — compile-verified
//
#include <hip/hip_runtime.h>
#include <math.h>

// Problem constants (match reference)
#define BATCH 8
#define CCH   256
#define HWPIX 1024          // 32*32
#define NHEAD 8
#define HDIM  32
#define CMID  128           // C/2 offset-net hidden channels
#define KCONV 2304          // 256*3*3

typedef __attribute__((ext_vector_type(2))) float v2f;
typedef __attribute__((ext_vector_type(8))) float v8f;

static __device__ __forceinline__ v8f wmma_f32(v2f a, v2f b, v8f c) {
  // V_WMMA_F32_16X16X4_F32 : D(16x16 f32) = A(16x4 f32) x B(4x16 f32) + C
  return __builtin_amdgcn_wmma_f32_16x16x4_f32(
      false, a, false, b, (short)0, c, false, false);
}

// ---------------------------------------------------------------------------
// fp32 WMMA GEMM (compile-time shape):  C[M x N] = A[M x K] * B[K x N] (+bias)
// One wave -> 32(M) x 64(N) tile, software-pipelined (prefetch next k-step).
// blockDim = 128 (4 waves). grid.x = (M/32)*(N/64)/4, grid.y = batch.
// ---------------------------------------------------------------------------
template <int M, int N, int K, bool HAS_BIAS>
__global__ __launch_bounds__(128) void gemm_wmma_f32(
    const float* __restrict__ A, const float* __restrict__ Bm,
    float* __restrict__ Cm, const float* __restrict__ bias) {
  const int lane = threadIdx.x & 31;
  const int wv   = threadIdx.x >> 5;
  constexpr int TILES_N = N >> 6;
  const int tile = blockIdx.x * 4 + wv;
  const int mt = tile / TILES_N;            // 32-row block
  const int nb = tile % TILES_N;            // 64-col block

  const float* Bp = Bm + (long)blockIdx.y * (K * N);
  float*       Cp = Cm + (long)blockIdx.y * (M * N);

  const int m16 = lane & 15;
  const int g   = lane >> 4;
  const int kg  = g << 1;                   // this lane-half covers K = kg,kg+1
  const float* Arow0 = A + (mt * 32 + m16) * K + kg;
  const float* Arow1 = Arow0 + 16 * K;
  const int n0 = nb * 64 + m16;

  v8f c00 = {}, c01 = {}, c02 = {}, c03 = {};
  v8f c10 = {}, c11 = {}, c12 = {}, c13 = {};

  // prologue: fragments for k-step 0
  v2f a0 = *(const v2f*)(Arow0);
  v2f a1 = *(const v2f*)(Arow1);
  const float* r0 = Bp + kg * N + n0;       // row k; row k+1 at +N (imm offset)
  v2f b0, b1, b2, b3;
  b0.x = r0[0];  b0.y = r0[N];
  b1.x = r0[16]; b1.y = r0[N + 16];
  b2.x = r0[32]; b2.y = r0[N + 32];
  b3.x = r0[48]; b3.y = r0[N + 48];

  for (int k0 = 4; k0 < K; k0 += 4) {
    // prefetch next k-step while current WMMAs execute
    v2f na0 = *(const v2f*)(Arow0 + k0);
    v2f na1 = *(const v2f*)(Arow1 + k0);
    const float* nr0 = Bp + (k0 + kg) * N + n0;
    v2f nb0, nb1, nb2, nb3;
    nb0.x = nr0[0];  nb0.y = nr0[N];
    nb1.x = nr0[16]; nb1.y = nr0[N + 16];
    nb2.x = nr0[32]; nb2.y = nr0[N + 32];
    nb3.x = nr0[48]; nb3.y = nr0[N + 48];

    c00 = wmma_f32(a0, b0, c00);  c10 = wmma_f32(a1, b0, c10);
    c01 = wmma_f32(a0, b1, c01);  c11 = wmma_f32(a1, b1, c11);
    c02 = wmma_f32(a0, b2, c02);  c12 = wmma_f32(a1, b2, c12);
    c03 = wmma_f32(a0, b3, c03);  c13 = wmma_f32(a1, b3, c13);

    a0 = na0; a1 = na1; b0 = nb0; b1 = nb1; b2 = nb2; b3 = nb3;
  }
  c00 = wmma_f32(a0, b0, c00);  c10 = wmma_f32(a1, b0, c10);
  c01 = wmma_f32(a0, b1, c01);  c11 = wmma_f32(a1, b1, c11);
  c02 = wmma_f32(a0, b2, c02);  c12 = wmma_f32(a1, b2, c12);
  c03 = wmma_f32(a0, b3, c03);  c13 = wmma_f32(a1, b3, c13);

#define STORE_ACC(ACC, T, J)                                                 \
  {                                                                          \
    const int col = nb * 64 + (J) * 16 + m16;                                \
    _Pragma("unroll") for (int r = 0; r < 8; ++r) {                          \
      const int mm = mt * 32 + (T) * 16 + r + 8 * g;                         \
      float val = ACC[r];                                                    \
      if (HAS_BIAS) val += bias[mm];                                         \
      Cp[mm * N + col] = val;                                                \
    }                                                                        \
  }
  STORE_ACC(c00, 0, 0) STORE_ACC(c01, 0, 1) STORE_ACC(c02, 0, 2) STORE_ACC(c03, 0, 3)
  STORE_ACC(c10, 1, 0) STORE_ACC(c11, 1, 1) STORE_ACC(c12, 1, 2) STORE_ACC(c13, 1, 3)
#undef STORE_ACC
}

// ---------------------------------------------------------------------------
// Repack Woff1 (128 x [c*9+j]) -> tap-major (128 x [j*256+c]). One-time tiny.
// ---------------------------------------------------------------------------
__global__ void repack_woff1(const float* __restrict__ Wsrc,
                             float* __restrict__ Wdst) {
  const int i = blockIdx.x * blockDim.x + threadIdx.x;   // 128*2304
  if (i >= CMID * KCONV) return;
  const int o = i / KCONV, k = i - o * KCONV;            // k = c*9 + j
  const int c = k / 9, j = k - c * 9;
  Wdst[o * KCONV + j * CCH + c] = Wsrc[i];
}

// ---------------------------------------------------------------------------
// Implicit-GEMM 3x3 SAME conv via WMMA, tap-major weights:
//   H1[b][128 x 1024] = Wp[128 x (9*256)] * im2col_tapmajor(Q[b][256 x 1024])
// Outer loop: 9 taps (rolled; per-tap mask+offset hoisted -> branch-free
// masked loads). Inner loop: 256 channels, pure strided addrs, pipelined.
// Loops pinned with unroll(1) to avoid register-pressure blowup.
// One wave -> 32(M) x 64(N). grid.x = (128/32)*16/4 = 16, grid.y = B.
// ---------------------------------------------------------------------------
__global__ __launch_bounds__(128) void conv3x3_wmma_f32(
    const float* __restrict__ Wp, const float* __restrict__ Q,
    float* __restrict__ H1) {
  const int lane = threadIdx.x & 31;
  const int wv   = threadIdx.x >> 5;
  const int tile = blockIdx.x * 4 + wv;          // 64 tiles: 4 (M/32) x 16 (N/64)
  const int mt = tile >> 4;
  const int nb = tile & 15;

  const float* Qb = Q  + (long)blockIdx.y * (CCH * HWPIX);
  float*       Hb = H1 + (long)blockIdx.y * (CMID * HWPIX);

  const int m16 = lane & 15;
  const int g   = lane >> 4;
  const int kg  = g << 1;
  const float* Arow0 = Wp + (mt * 32 + m16) * KCONV + kg;
  const float* Arow1 = Arow0 + 16 * KCONV;
  const int nbase = nb * 64 + m16;

  v8f c00 = {}, c01 = {}, c02 = {}, c03 = {};
  v8f c10 = {}, c11 = {}, c12 = {}, c13 = {};

#pragma unroll 1
  for (int j = 0; j < 9; ++j) {
    const int ky = j / 3 - 1, kx = j - (j / 3) * 3 - 1;  // scalar, once per tap
    // per-column-tile offset + mask: invariant over the channel loop
    int   off[4];
    float msk[4];
#pragma unroll
    for (int t = 0; t < 4; ++t) {
      const int p = nbase + t * 16;
      const int y = p >> 5, x = p & 31;
      const int sy = y + ky, sx = x + kx;
      const bool ok = ((unsigned)sy < 32u) && ((unsigned)sx < 32u);
      msk[t] = ok ? 1.f : 0.f;
      off[t] = ok ? (sy * 32 + sx) : 0;
    }
    const float* Aj0 = Arow0 + j * CCH;
    const float* Aj1 = Arow1 + j * CCH;

    // prologue: channel step 0
    v2f a0 = *(const v2f*)(Aj0);
    v2f a1 = *(const v2f*)(Aj1);
    const float* qc = Qb + kg * HWPIX;
    v2f b0, b1, b2, b3;
    b0.x = msk[0] * qc[off[0]]; b0.y = msk[0] * qc[HWPIX + off[0]];
    b1.x = msk[1] * qc[off[1]]; b1.y = msk[1] * qc[HWPIX + off[1]];
    b2.x = msk[2] * qc[off[2]]; b2.y = msk[2] * qc[HWPIX + off[2]];
    b3.x = msk[3] * qc[off[3]]; b3.y = msk[3] * qc[HWPIX + off[3]];

#pragma unroll 1
    for (int c = 4; c < CCH; c += 4) {
      v2f na0 = *(const v2f*)(Aj0 + c);
      v2f na1 = *(const v2f*)(Aj1 + c);
      const float* nqc = Qb + (c + kg) * HWPIX;
      v2f nb0, nb1, nb2, nb3;
      nb0.x = msk[0] * nqc[off[0]]; nb0.y = msk[0] * nqc[HWPIX + off[0]];
      nb1.x = msk[1] * nqc[off[1]]; nb1.y = msk[1] * nqc[HWPIX + off[1]];
      nb2.x = msk[2] * nqc[off[2]]; nb2.y = msk[2] * nqc[HWPIX + off[2]];
      nb3.x = msk[3] * nqc[off[3]]; nb3.y = msk[3] * nqc[HWPIX + off[3]];

      c00 = wmma_f32(a0, b0, c00);  c10 = wmma_f32(a1, b0, c10);
      c01 = wmma_f32(a0, b1, c01);  c11 = wmma_f32(a1, b1, c11);
      c02 = wmma_f32(a0, b2, c02);  c12 = wmma_f32(a1, b2, c12);
      c03 = wmma_f32(a0, b3, c03);  c13 = wmma_f32(a1, b3, c13);

      a0 = na0; a1 = na1; b0 = nb0; b1 = nb1; b2 = nb2; b3 = nb3;
    }
    c00 = wmma_f32(a0, b0, c00);  c10 = wmma_f32(a1, b0, c10);
    c01 = wmma_f32(a0, b1, c01);  c11 = wmma_f32(a1, b1, c11);
    c02 = wmma_f32(a0, b2, c02);  c12 = wmma_f32(a1, b2, c12);
    c03 = wmma_f32(a0, b3, c03);  c13 = wmma_f32(a1, b3, c13);
  }

#define STORE_ACC(ACC, T, J)                                                 \
  {                                                                          \
    const int col = nb * 64 + (J) * 16 + m16;                                \
    _Pragma("unroll") for (int r = 0; r < 8; ++r) {                          \
      const int mm = mt * 32 + (T) * 16 + r + 8 * g;                         \
      Hb[mm * HWPIX + col] = ACC[r];                                         \
    }                                                                        \
  }
  STORE_ACC(c00, 0, 0) STORE_ACC(c01, 0, 1) STORE_ACC(c02, 0, 2) STORE_ACC(c03, 0, 3)
  STORE_ACC(c10, 1, 0) STORE_ACC(c11, 1, 1) STORE_ACC(c12, 1, 2) STORE_ACC(c13, 1, 3)
#undef STORE_ACC
}

// ---------------------------------------------------------------------------
// Offset head: bias + ReLU + 2-row 1x1 conv (reference uses only offset
// group 0 -> Woff2 rows 0,1), then bilinear sample setup (border clamp,
// align_corners). One thread per (b, pixel).
// ---------------------------------------------------------------------------
__global__ void offsets_kernel(const float* __restrict__ H1,
                               const float* __restrict__ boff1,
                               const float* __restrict__ W2,
                               const float* __restrict__ boff2,
                               int* __restrict__ sampIdx,
                               float* __restrict__ sampW) {
  const int i = blockIdx.x * blockDim.x + threadIdx.x;   // B*HW
  if (i >= BATCH * HWPIX) return;
  const int b = i >> 10, p = i & 1023;
  const float* h = H1 + (long)b * CMID * HWPIX + p;
  float o0 = 0.f, o1 = 0.f;
  for (int c = 0; c < CMID; ++c) {
    float t = h[c * HWPIX] + boff1[c];
    t = t > 0.f ? t : 0.f;
    o0 = fmaf(W2[c], t, o0);
    o1 = fmaf(W2[CMID + c], t, o1);
  }
  const float offx = (o0 + boff2[0]) * 0.1f;
  const float offy = (o1 + boff2[1]) * 0.1f;
  const int x = p & 31, y = p >> 5;
  const float vx = (2.f * x) / 31.f - 1.f + offx;
  const float vy = (2.f * y) / 31.f - 1.f + offy;
  float xf = (vx + 1.f) * 0.5f * 31.f;
  float yf = (vy + 1.f) * 0.5f * 31.f;
  xf = fminf(fmaxf(xf, 0.f), 31.f);
  yf = fminf(fmaxf(yf, 0.f), 31.f);
  const float x0 = floorf(xf), y0 = floorf(yf);
  const float wx = xf - x0, wy = yf - y0;
  const int x0i = (int)x0, y0i = (int)y0;
  const int x1i = min(x0i + 1, 31), y1i = min(y0i + 1, 31);
  const int base = i * 4;
  sampIdx[base + 0] = y0i * 32 + x0i;
  sampIdx[base + 1] = y0i * 32 + x1i;
  sampIdx[base + 2] = y1i * 32 + x0i;
  sampIdx[base + 3] = y1i * 32 + x1i;
  sampW[base + 0] = (1.f - wx) * (1.f - wy);
  sampW[base + 1] = wx * (1.f - wy);
  sampW[base + 2] = (1.f - wx) * wy;
  sampW[base + 3] = wx * wy;
}

// ---------------------------------------------------------------------------
// Bilinear gather: kv_sampled[b][c][p] from kv_map with per-(b,p) idx/weights.
// ---------------------------------------------------------------------------
__global__ void gather_kernel(const float* __restrict__ kv,
                              const int* __restrict__ sampIdx,
                              const float* __restrict__ sampW,
                              float* __restrict__ out) {
  const long i = (long)blockIdx.x * blockDim.x + threadIdx.x;  // B*C*HW
  if (i >= (long)BATCH * CCH * HWPIX) return;
  const int p  = (int)(i & 1023);
  const long bc = i >> 10;
  const int b  = (int)(bc >> 8);
  const float* src = kv + bc * HWPIX;
  const int s = (b * HWPIX + p) * 4;
  const int4   idx = *(const int4*)(sampIdx + s);
  const float4 w   = *(const float4*)(sampW + s);
  out[i] = w.x * src[idx.x] + w.y * src[idx.y] +
           w.z * src[idx.z] + w.w * src[idx.w];
}

// ---------------------------------------------------------------------------
// Attention: one wave per (b, head, 16-query tile). Scores via WMMA into a
// 16x1024 LDS buffer (64 KB), softmax in LDS (2 lanes/row, float4), then
// P * V^T via WMMA with A-frags from LDS. grid.x = B*NHEAD*64, blockDim = 32.
// ---------------------------------------------------------------------------
__global__ __launch_bounds__(32) void attn_kernel(
    const float* __restrict__ Qb, const float* __restrict__ Kb,
    const float* __restrict__ Vb, float* __restrict__ Ob) {
  __shared__ float sc[16 * HWPIX];                 // 65536 B
  const int lane = threadIdx.x;
  const int blk = blockIdx.x;
  const int qt = blk & 63;
  const int hh = (blk >> 6) & (NHEAD - 1);
  const int b  = blk >> 9;
  const long base = ((long)b * CCH + hh * HDIM) * HWPIX;
  const float* q = Qb + base;
  const float* k = Kb + base;
  const float* v = Vb + base;
  float*       o = Ob + base;
  const int q0 = qt * 16;
  const int m16 = lane & 15;
  const int g   = lane >> 4;
  const int kg  = g << 1;
  const float scale = 0.17677669529663687f;        // 32^-0.5

  // Pass 1: scores[16 x 1024] = (q tile)^T * k, scaled.
  // q A-frags invariant across key tiles: hoist all 8.
  v2f aq[8];
#pragma unroll
  for (int i = 0; i < 8; ++i) {
    aq[i].x = q[(4 * i + kg) * HWPIX + q0 + m16];
    aq[i].y = q[(4 * i + kg + 1) * HWPIX + q0 + m16];
  }
#pragma unroll 1
  for (int kt = 0; kt < 64; ++kt) {
    v8f s0 = {}, s1 = {};
    const float* kp = k + kt * 16 + m16;
#pragma unroll
    for (int i = 0; i < 4; ++i) {
      v2f b0, b1;
      b0.x = kp[(8 * i + kg) * HWPIX];
      b0.y = kp[(8 * i + kg + 1) * HWPIX];
      b1.x = kp[(8 * i + 4 + kg) * HWPIX];
      b1.y = kp[(8 * i + 5 + kg) * HWPIX];
      s0 = wmma_f32(aq[2 * i], b0, s0);
      s1 = wmma_f32(aq[2 * i + 1], b1, s1);
    }
#pragma unroll
    for (int r = 0; r < 8; ++r)
      sc[(r + 8 * g) * HWPIX + kt * 16 + m16] = (s0[r] + s1[r]) * scale;
  }
  __syncthreads();

  // Pass 2: row softmax, 2 lanes per row, float4 LDS traffic.
  {
    float* rp = sc + m16 * HWPIX + g * 512;
    float mx = -3.4e38f;
    for (int i = 0; i < 512; i += 4) {
      const float4 t = *(const float4*)(rp + i);
      mx = fmaxf(mx, fmaxf(fmaxf(t.x, t.y), fmaxf(t.z, t.w)));
    }
    mx = fmaxf(mx, __shfl_xor(mx, 16, 32));
    float sum = 0.f;
    for (int i = 0; i < 512; i += 4) {
      float4 t = *(const float4*)(rp + i);
      t.x = __expf(t.x - mx); t.y = __expf(t.y - mx);
      t.z = __expf(t.z - mx); t.w = __expf(t.w - mx);
      sum += (t.x + t.y) + (t.z + t.w);
      *(float4*)(rp + i) = t;
    }
    sum += __shfl_xor(sum, 16, 32);
    const float inv = 1.f / sum;
    for (int i = 0; i < 512; i += 4) {
      float4 t = *(const float4*)(rp + i);
      t.x *= inv; t.y *= inv; t.z *= inv; t.w *= inv;
      *(float4*)(rp + i) = t;
    }
  }
  __syncthreads();

  // Pass 3: out[16 q x 32 d] = P(16x1024) * V^T(1024x32), 4 accumulators.
  v8f o00 = {}, o01 = {}, o10 = {}, o11 = {};
  const float* srow = sc + m16 * HWPIX + kg;
  const float* v0   = v + m16 * HWPIX + kg;
  const float* v1   = v + (16 + m16) * HWPIX + kg;
#pragma unroll 1
  for (int k0 = 0; k0 < HWPIX; k0 += 8) {
    const v2f aA = *(const v2f*)(srow + k0);
    const v2f aB = *(const v2f*)(srow + k0 + 4);
    const v2f bA0 = *(const v2f*)(v0 + k0);
    const v2f bB0 = *(const v2f*)(v0 + k0 + 4);
    const v2f bA1 = *(const v2f*)(v1 + k0);
    const v2f bB1 = *(const v2f*)(v1 + k0 + 4);
    o00 = wmma_f32(aA, bA0, o00);
    o10 = wmma_f32(aA, bA1, o10);
    o01 = wmma_f32(aB, bB0, o01);
    o11 = wmma_f32(aB, bB1, o11);
  }
  // store transposed back to (C, HW): channel from n(lane), pixel from m(vgpr)
  float* dst0 = o + m16 * HWPIX + q0 + 8 * g;
  float* dst1 = o + (16 + m16) * HWPIX + q0 + 8 * g;
#pragma unroll
  for (int r = 0; r < 8; ++r) {
    dst0[r] = o00[r] + o01[r];
    dst1[r] = o10[r] + o11[r];
  }
}

// ---------------------------------------------------------------------------
extern "C" void kernel_launch(void* const* d_in, const int* in_sizes, int n_in,
                              void* d_out, int out_size, void* d_ws, size_t ws_size,
                              hipStream_t stream) {
  const float* query_map = (const float*)d_in[0];
  const float* kv_map    = (const float*)d_in[1];
  const float* Wq        = (const float*)d_in[2];
  const float* Wk        = (const float*)d_in[3];
  const float* Wv        = (const float*)d_in[4];
  const float* Woff1     = (const float*)d_in[5];
  const float* boff1     = (const float*)d_in[6];
  const float* Woff2     = (const float*)d_in[7];
  const float* boff2     = (const float*)d_in[8];
  const float* Wout      = (const float*)d_in[9];
  const float* bout      = (const float*)d_in[10];
  float* out = (float*)d_out;

  // Workspace layout (floats)
  const long NQ  = (long)BATCH * CCH * HWPIX;        // 2,097,152
  const long NH1 = (long)BATCH * CMID * HWPIX;       // 1,048,576
  const long NS  = (long)BATCH * HWPIX * 4;          // 32,768
  const long NWP = (long)CMID * KCONV;               // 294,912
  float* ws   = (float*)d_ws;
  float* qbuf = ws;                       // NQ
  float* h1   = qbuf + NQ;                // NH1
  int*   sIdx = (int*)(h1 + NH1);         // NS ints
  float* sW   = (float*)(sIdx + NS);      // NS
  float* kvs  = sW + NS;                  // NQ (reused as attn-out after K/V proj)
  float* kbuf = kvs + NQ;                 // NQ
  float* vbuf = kbuf + NQ;                // NQ
  float* wprm = vbuf + NQ;                // NWP (tap-major conv weights)
  float* aout = kvs;                      // alias: kvs dead after K/V projections

  // 0. repack conv weights tap-major
  repack_woff1<<<(int)((NWP + 255) / 256), 256, 0, stream>>>(Woff1, wprm);
  // 1. q = Wq * query_map          (M=256, N=1024, K=256)
  gemm_wmma_f32<CCH, HWPIX, CCH, false><<<dim3(32, BATCH), 128, 0, stream>>>(
      Wq, query_map, qbuf, nullptr);
  // 2. h1 = conv3x3(q, Woff1)      (implicit GEMM, tap-major)
  conv3x3_wmma_f32<<<dim3(16, BATCH), 128, 0, stream>>>(wprm, qbuf, h1);
  // 3. offsets + bilinear sample setup
  offsets_kernel<<<(BATCH * HWPIX + 255) / 256, 256, 0, stream>>>(
      h1, boff1, Woff2, boff2, sIdx, sW);
  // 4. kv_sampled = grid_sample(kv_map)
  gather_kernel<<<(int)((NQ + 255) / 256), 256, 0, stream>>>(kv_map, sIdx, sW, kvs);
  // 5. k = Wk * kv_sampled ; 6. v = Wv * kv_sampled
  gemm_wmma_f32<CCH, HWPIX, CCH, false><<<dim3(32, BATCH), 128, 0, stream>>>(
      Wk, kvs, kbuf, nullptr);
  gemm_wmma_f32<CCH, HWPIX, CCH, false><<<dim3(32, BATCH), 128, 0, stream>>>(
      Wv, kvs, vbuf, nullptr);
  // 7. attention (LDS score buffer flash-lite)
  attn_kernel<<<BATCH * NHEAD * 64, 32, 0, stream>>>(qbuf, kbuf, vbuf, aout);
  // 8. out = Wout * attn_out + bout
  gemm_wmma_f32<CCH, HWPIX, CCH, true><<<dim3(32, BATCH), 128, 0, stream>>>(
      Wout, aout, out, bout);
}